// DetectionLoss_57088705298996
// MI455X (gfx1250) — compile-verified
//
#include <hip/hip_runtime.h>
#include <hip/hip_bf16.h>
#include <math.h>

#define AN 32768
#define BI 32
#define GTN 64
#define IOU_TH 0.5f

typedef float v2f __attribute__((ext_vector_type(2)));
typedef float v8f __attribute__((ext_vector_type(8)));

// ---------- CDNA5 async-to-LDS helpers (inline asm, ASYNCcnt tracked) ----------
__device__ __forceinline__ uint32_t lds_off_u32(const void* p) {
  // generic address of a __shared__ object is {SHARED_BASE[63:32], lds_offset[31:0]}
  return (uint32_t)(uintptr_t)p;
}
__device__ __forceinline__ void async_b128_to_lds(uint32_t lds, const void* g) {
  asm volatile("global_load_async_to_lds_b128 %0, %1, off" :: "v"(lds), "v"(g) : "memory");
}
__device__ __forceinline__ void async_b32_to_lds(uint32_t lds, const void* g) {
  asm volatile("global_load_async_to_lds_b32 %0, %1, off" :: "v"(lds), "v"(g) : "memory");
}
__device__ __forceinline__ void wait_async0() {
  asm volatile("s_wait_asynccnt 0" ::: "memory");
}

// ---------- wave32 reductions ----------
__device__ __forceinline__ float waveSumF(float v) {
  for (int o = 16; o > 0; o >>= 1) v += __shfl_xor(v, o);
  return v;
}
__device__ __forceinline__ int waveSumI(int v) {
  for (int o = 16; o > 0; o >>= 1) v += __shfl_xor(v, o);
  return v;
}
__device__ __forceinline__ unsigned long long shflxor_u64(unsigned long long v, int m) {
  uint32_t lo = (uint32_t)v, hi = (uint32_t)(v >> 32);
  lo = __shfl_xor(lo, m); hi = __shfl_xor(hi, m);
  return ((unsigned long long)hi << 32) | lo;
}

__device__ __forceinline__ float boxArea(float4 b) { return (b.z - b.x) * (b.w - b.y); }
__device__ __forceinline__ float sL1(float d) {
  float ad = fabsf(d);
  return ad < 1.0f ? 0.5f * d * d : ad - 0.5f;
}

// ---------- K0: zero accumulators ----------
__global__ void init_kernel(float* locSum, float* posConf, int* numPos) {
  int t = threadIdx.x;
  if (t < BI) { locSum[t] = 0.0f; posConf[t] = 0.0f; numPos[t] = 0; }
}

// ---------- K1: per-anchor best GT (max + first-argmax over G) ----------
__global__ void __launch_bounds__(256)
match_kernel(const float4* __restrict__ anchors, const float4* __restrict__ gts,
             int* __restrict__ combined) {
  __shared__ float4 sGt[GTN];
  const int b = blockIdx.y;
  const int tid = threadIdx.x;
  if (tid < GTN)
    async_b128_to_lds(lds_off_u32(&sGt[tid]), (const void*)(gts + (size_t)b * GTN + tid));
  wait_async0();
  __syncthreads();

  const int a = blockIdx.x * 256 + tid;
  float4 an = anchors[a];
  float a1 = boxArea(an);
  float best = -1.0f; int bidx = 0;
#pragma unroll 8
  for (int g = 0; g < GTN; ++g) {
    float4 gb = sGt[g];
    float lx = fmaxf(an.x, gb.x), ly = fmaxf(an.y, gb.y);
    float rx = fminf(an.z, gb.z), ry = fminf(an.w, gb.w);
    float w = fmaxf(rx - lx, 0.0f), h = fmaxf(ry - ly, 0.0f);
    float inter = w * h;
    float v = inter / (a1 + boxArea(gb) - inter + 1e-6f);
    if (v > best) { best = v; bidx = g; }   // strict > keeps first max (jnp.argmax)
  }
  combined[(size_t)b * AN + a] = bidx | ((best > IOU_TH) ? (int)0x80000000u : 0);
}

// ---------- K1b: per-GT best anchor (first-argmax over A) ----------
__global__ void __launch_bounds__(256)
best_anchor_kernel(const float4* __restrict__ anchors, const float4* __restrict__ gts,
                   int* __restrict__ bestAIdx) {
  const int b = blockIdx.y;
  const int g0 = blockIdx.x * 8;
  const int tid = threadIdx.x;
  float4 G4[8]; float A2[8];
#pragma unroll
  for (int i = 0; i < 8; ++i) { G4[i] = gts[(size_t)b * GTN + g0 + i]; A2[i] = boxArea(G4[i]); }
  unsigned long long key[8];
#pragma unroll
  for (int i = 0; i < 8; ++i) key[i] = 0ull;

  for (int a = tid; a < AN; a += 256) {
    float4 an = anchors[a];
    float a1 = boxArea(an);
    unsigned long long alow = (unsigned long long)(uint32_t)(~a); // smaller a -> larger key
#pragma unroll
    for (int i = 0; i < 8; ++i) {
      float lx = fmaxf(an.x, G4[i].x), ly = fmaxf(an.y, G4[i].y);
      float rx = fminf(an.z, G4[i].z), ry = fminf(an.w, G4[i].w);
      float w = fmaxf(rx - lx, 0.0f), h = fmaxf(ry - ly, 0.0f);
      float inter = w * h;
      float v = inter / (a1 + A2[i] - inter + 1e-6f);
      unsigned long long k = ((unsigned long long)__float_as_uint(v) << 32) | alow;
      key[i] = (k > key[i]) ? k : key[i];
    }
  }
  __shared__ unsigned long long sB[8];
  if (tid < 8) sB[tid] = 0ull;
  __syncthreads();
  const int lane = tid & 31;
#pragma unroll
  for (int i = 0; i < 8; ++i) {
    unsigned long long k = key[i];
    for (int o = 16; o > 0; o >>= 1) {
      unsigned long long other = shflxor_u64(k, o);
      k = (other > k) ? other : k;
    }
    if (lane == 0) atomicMax(&sB[i], k);   // ds_max_u64
  }
  __syncthreads();
  if (tid < 8) bestAIdx[(size_t)b * GTN + g0 + tid] = (int)(~(uint32_t)sB[tid]);
}

// ---------- K2: positives, loc loss, positive BCE, num_pos ----------
__global__ void __launch_bounds__(256)
loss_kernel(const float4* __restrict__ bbox, const float* __restrict__ conf,
            const float4* __restrict__ gts, const int* __restrict__ bestAIdx,
            int* __restrict__ combined, float* __restrict__ locSum,
            float* __restrict__ posConf, int* __restrict__ numPos) {
  __shared__ float4 sGt[GTN];
  __shared__ int sIds[GTN];
  __shared__ float sLoc, sPC; __shared__ int sNp;
  const int b = blockIdx.y;
  const int tid = threadIdx.x;
  if (tid < GTN) {
    async_b128_to_lds(lds_off_u32(&sGt[tid]), (const void*)(gts + (size_t)b * GTN + tid));
    async_b32_to_lds(lds_off_u32(&sIds[tid]), (const void*)(bestAIdx + (size_t)b * GTN + tid));
  }
  if (tid == 0) { sLoc = 0.0f; sPC = 0.0f; sNp = 0; }
  wait_async0();
  __syncthreads();

  const int a = blockIdx.x * 256 + tid;
  const size_t off = (size_t)b * AN + a;
  int w = combined[off];
  int idx = w & (GTN - 1);
  bool pos = ((unsigned)w >> 31) != 0;
#pragma unroll 8
  for (int g = 0; g < GTN; ++g) pos = pos || (sIds[g] == a);  // forced positives

  float4 p4 = bbox[off];
  float4 m4 = sGt[idx];
  float d0 = 0.5f * (p4.x + p4.z) - 0.5f * (m4.x + m4.z);
  float d1 = 0.5f * (p4.y + p4.w) - 0.5f * (m4.y + m4.w);
  float d2 = (p4.z - p4.x) - (m4.z - m4.x);
  float d3 = (p4.w - p4.y) - (m4.w - m4.y);
  float sl = sL1(d0) + sL1(d1) + sL1(d2) + sL1(d3);

  float p = conf[off];
  float bcePos = fminf(-logf(p), 100.0f);           // -clip(log p, -100)

  float locC = pos ? sl : 0.0f;
  float pcC  = pos ? bcePos : 0.0f;
  int   npC  = pos ? 1 : 0;
  combined[off] = idx | (pos ? (int)0x80000000u : 0);  // final pos bit for K3

  locC = waveSumF(locC); pcC = waveSumF(pcC); npC = waveSumI(npC);
  if ((tid & 31) == 0) { atomicAdd(&sLoc, locC); atomicAdd(&sPC, pcC); atomicAdd(&sNp, npC); }
  __syncthreads();
  if (tid == 0) { atomicAdd(&locSum[b], sLoc); atomicAdd(&posConf[b], sPC); atomicAdd(&numPos[b], sNp); }
}

// ---------- K3: hard-negative mining (exact top-K sum via 32-bit radix select) ----------
__global__ void __launch_bounds__(1024)
hardneg_kernel(const int* __restrict__ combined, const float* __restrict__ conf,
               const int* __restrict__ numPos, float* __restrict__ negConf) {
  const int b = blockIdx.x;
  const int tid = threadIdx.x;
  const int lane = tid & 31;
  const int np = numPos[b];
  int K = np * 3;
  if (K > AN - np) K = AN - np;
  if (K <= 0) { if (tid == 0) negConf[b] = 0.0f; return; }   // uniform branch

  uint32_t r[32];                                   // all 32768 values live in registers
#pragma unroll
  for (int i = 0; i < 32; ++i) {
    int a = i * 1024 + tid;
    size_t off = (size_t)b * AN + a;
    bool pos = ((unsigned)combined[off] >> 31) != 0;
    float p = conf[off];
    float v = pos ? 0.0f : fminf(-log1pf(-p), 100.0f);   // bce(t=0), >= 0
    r[i] = __float_as_uint(v);                            // order-preserving (v >= 0)
  }

  __shared__ int sCnt;
  __shared__ float sSum;
  uint32_t prefix = 0;
  for (int bit = 31; bit >= 0; --bit) {             // find K-th largest value exactly
    uint32_t cand = prefix | (1u << bit);
    int c = 0;
#pragma unroll
    for (int i = 0; i < 32; ++i) c += (r[i] >= cand) ? 1 : 0;
    c = waveSumI(c);
    if (tid == 0) sCnt = 0;
    __syncthreads();
    if (lane == 0) atomicAdd(&sCnt, c);
    __syncthreads();
    if (sCnt >= K) prefix = cand;
    __syncthreads();
  }
  int cg = 0; float s = 0.0f;
#pragma unroll
  for (int i = 0; i < 32; ++i)
    if (r[i] > prefix) { cg++; s += __uint_as_float(r[i]); }
  cg = waveSumI(cg); s = waveSumF(s);
  if (tid == 0) { sCnt = 0; sSum = 0.0f; }
  __syncthreads();
  if (lane == 0) { atomicAdd(&sCnt, cg); atomicAdd(&sSum, s); }
  __syncthreads();
  if (tid == 0)
    negConf[b] = sSum + (float)(K - sCnt) * __uint_as_float(prefix); // ties at threshold
}

// ---------- K4: finalize with WMMA 32-lane reduction (EXEC all-1s: one full wave) ----------
__device__ __forceinline__ float wmmaWaveSum(float v) {
  // A(16x4) = {v,0} per lane -> rowsum(m) = v[m] + v[m+16]; B(4x16) = ones; C = 0.
  v2f a; a.x = v; a.y = 0.0f;
  v2f bb; bb.x = 1.0f; bb.y = 1.0f;
  v8f c = {};
  v8f d = __builtin_amdgcn_wmma_f32_16x16x4_f32(false, a, false, bb, (short)0, c, false, false);
  float s = d[0] + d[1] + d[2] + d[3] + d[4] + d[5] + d[6] + d[7];
  s += __shfl_xor(s, 16);   // combine the two half-wave row-sum groups
  return s;
}

__global__ void __launch_bounds__(32)
finalize_kernel(const float* __restrict__ locSum, const float* __restrict__ posConf,
                const float* __restrict__ negConf, const int* __restrict__ numPos,
                float* __restrict__ out) {
  int b = threadIdx.x;                 // 0..31, exactly B images, all lanes active
  float loc  = locSum[b];
  float cf   = posConf[b] + negConf[b];
  float np   = (float)numPos[b];
  float locT = wmmaWaveSum(loc);
  float cfT  = wmmaWaveSum(cf);
  float npT  = wmmaWaveSum(np);
  float tp = fmaxf(1.0f, npT);
  if (b == 0) {
    float locL = locT / tp, cfL = cfT / tp;
    out[0] = locL + cfL; out[1] = cfL; out[2] = locL;
  }
}

extern "C" void kernel_launch(void* const* d_in, const int* in_sizes, int n_in,
                              void* d_out, int out_size, void* d_ws, size_t ws_size,
                              hipStream_t stream) {
  const float4* bbox    = (const float4*)d_in[0];  // [B,A,4]
  const float*  conf    = (const float*) d_in[1];  // [B,A]
  const float4* anchors = (const float4*)d_in[2];  // [A,4]
  const float4* gts     = (const float4*)d_in[3];  // [B,G,4]

  char* ws = (char*)d_ws;
  int* combined = (int*)ws;                                          // B*A ints (4 MB)
  int* bestAIdx = (int*)(ws + (size_t)BI * AN * sizeof(int));        // B*G ints
  uintptr_t tail = (uintptr_t)(ws + (size_t)BI * AN * sizeof(int) + (size_t)BI * GTN * sizeof(int));
  tail = (tail + 255) & ~(uintptr_t)255;
  float* locSum  = (float*)tail;
  float* posConf = locSum + BI;
  float* negConf = posConf + BI;
  int*   numPos  = (int*)(negConf + BI);
  float* out = (float*)d_out;

  init_kernel<<<dim3(1), dim3(64), 0, stream>>>(locSum, posConf, numPos);
  match_kernel<<<dim3(AN / 256, BI), dim3(256), 0, stream>>>(anchors, gts, combined);
  best_anchor_kernel<<<dim3(GTN / 8, BI), dim3(256), 0, stream>>>(anchors, gts, bestAIdx);
  loss_kernel<<<dim3(AN / 256, BI), dim3(256), 0, stream>>>(bbox, conf, gts, bestAIdx,
                                                            combined, locSum, posConf, numPos);
  hardneg_kernel<<<dim3(BI), dim3(1024), 0, stream>>>(combined, conf, numPos, negConf);
  finalize_kernel<<<dim3(1), dim3(32), 0, stream>>>(locSum, posConf, negConf, numPos, out);
}